// LargeSDFSurface_78374563217924
// MI455X (gfx1250) — compile-verified
//
#include <hip/hip_runtime.h>
#include <math.h>

typedef __attribute__((ext_vector_type(2))) float v2f;
typedef __attribute__((ext_vector_type(4))) float v4f;
typedef __attribute__((ext_vector_type(8))) float v8f;

#define TILES_PER_WAVE 16
#define BLOCK_THREADS  256

// Min-reduce two independent values across each 16-lane half of the wave.
// ds_swizzle_b32 group-of-32 XOR butterfly (masks 1,2,4,8 stay inside each
// half). Two rows are interleaved so the first v_min_num issues under
// s_wait_dscnt 0x1 while the second swizzle is still in flight (DS ops
// return in order). Inline asm avoids the v_max canonicalize clang wraps
// around fminf on bitcast values.
__device__ __forceinline__ void min2_across16(float& x0, float& x1) {
    float t0, t1;
    asm volatile(
        "ds_swizzle_b32 %2, %0 offset:0x41f\n\t"   // xor 1
        "ds_swizzle_b32 %3, %1 offset:0x41f\n\t"
        "s_wait_dscnt 0x1\n\t"
        "v_min_num_f32 %0, %0, %2\n\t"
        "s_wait_dscnt 0x0\n\t"
        "v_min_num_f32 %1, %1, %3\n\t"
        "ds_swizzle_b32 %2, %0 offset:0x81f\n\t"   // xor 2
        "ds_swizzle_b32 %3, %1 offset:0x81f\n\t"
        "s_wait_dscnt 0x1\n\t"
        "v_min_num_f32 %0, %0, %2\n\t"
        "s_wait_dscnt 0x0\n\t"
        "v_min_num_f32 %1, %1, %3\n\t"
        "ds_swizzle_b32 %2, %0 offset:0x101f\n\t"  // xor 4
        "ds_swizzle_b32 %3, %1 offset:0x101f\n\t"
        "s_wait_dscnt 0x1\n\t"
        "v_min_num_f32 %0, %0, %2\n\t"
        "s_wait_dscnt 0x0\n\t"
        "v_min_num_f32 %1, %1, %3\n\t"
        "ds_swizzle_b32 %2, %0 offset:0x201f\n\t"  // xor 8
        "ds_swizzle_b32 %3, %1 offset:0x201f\n\t"
        "s_wait_dscnt 0x1\n\t"
        "v_min_num_f32 %0, %0, %2\n\t"
        "s_wait_dscnt 0x0\n\t"
        "v_min_num_f32 %1, %1, %3"
        : "+v"(x0), "+v"(x1), "=&v"(t0), "=&v"(t1));
}

// Single-instruction clamp to >= 0. v_max_num drops NaN, so d = -eps or NaN
// from WMMA rounding becomes 0 -> sqrt(0)-r = -r (the correct limit).
__device__ __forceinline__ float max0(float x) {
    float r;
    asm("v_max_num_f32 %0, 0, %1" : "=v"(r) : "v"(x));
    return r;
}

// One wave computes a 16-point x 32-sphere tile per iteration using
// V_WMMA_F32_16X16X4_F32:  D = A(16x4) * B(4x16) + C
//   A row m = [px, py, pz, ||p||^2]
//   B col n = [-2cx, -2cy, -2cz, 1]
//   C col n = ||c_n||^2  (lane-uniform per column -> broadcast into accumulator)
// => D[m][n] = ||p_m - c_n||^2. Then sdf = v_sqrt_f32(D) - r_n, min over the
// 32 sphere slots with the ds_swizzle butterfly above.
__global__ __launch_bounds__(BLOCK_THREADS)
void LargeSDFSurface_sdf_wmma_kernel(const float* __restrict__ points,
                                     const float* __restrict__ centers,
                                     const float* __restrict__ radii,
                                     float* __restrict__ out, int n)
{
    const int lane = threadIdx.x & 31;
    const int waveInBlock = threadIdx.x >> 5;
    const int wavesPerBlock = BLOCK_THREADS >> 5;
    // wave id is wave-uniform: pin it to an SGPR so loop control / bases are scalar
    const int gwave = __builtin_amdgcn_readfirstlane(
        blockIdx.x * wavesPerBlock + waveInBlock);
    const int col = lane & 15;   // column (sphere slot within tile) this lane owns
    const int hi  = lane >> 4;   // 0: K=0,1 provider ; 1: K=2,3 provider

    // ---- Loop-invariant B matrices / per-column constants -------------------
    // tile0 columns = spheres 0..15, tile1 columns = spheres 16..31 (31 = dummy)
    const float c0x = centers[3 * col + 0];
    const float c0y = centers[3 * col + 1];
    const float c0z = centers[3 * col + 2];
    const float r0  = radii[col];

    const int s1 = col + 16;
    float c1x = 0.0f, c1y = 0.0f, c1z = 0.0f;
    float r1  = -1.0e30f;               // dummy sphere: sdf ~ +1e30, never the min
    if (s1 < 31) {
        c1x = centers[3 * s1 + 0];
        c1y = centers[3 * s1 + 1];
        c1z = centers[3 * s1 + 2];
        r1  = radii[s1];
    }

    const float c2_0 = c0x * c0x + c0y * c0y + c0z * c0z;
    const float c2_1 = c1x * c1x + c1y * c1y + c1z * c1z;

    // B layout mirrors A 16x4: vgpr0 = K0 (lanes 0-15) / K2 (lanes 16-31),
    //                          vgpr1 = K1 / K3
    v2f b0, b1;
    b0[0] = hi ? (-2.0f * c0z) : (-2.0f * c0x);
    b0[1] = hi ? 1.0f          : (-2.0f * c0y);
    b1[0] = hi ? (-2.0f * c1z) : (-2.0f * c1x);
    b1[1] = hi ? 1.0f          : (-2.0f * c1y);

    // Accumulators hold ||c_n||^2 (all 8 slots of a lane belong to one column).
    v8f cacc0, cacc1;
#pragma unroll
    for (int v = 0; v < 8; ++v) { cacc0[v] = c2_0; cacc1[v] = c2_1; }

    const int waveBase = gwave * (16 * TILES_PER_WAVE);

    for (int t = 0; t < TILES_PER_WAVE; ++t) {
        const int base = waveBase + t * 16;            // SGPR
        if (base >= n) break;                          // scalar branch

        // ---- Load A: both half-waves read the same 16 points ---------------
        const int pidx = base + col;
        const unsigned pc = (unsigned)(pidx < n ? pidx : (n - 1)) * 3u;
        const float px = points[pc + 0];
        const float py = points[pc + 1];
        const float pz = points[pc + 2];

        v2f a;
        a[0] = hi ? pz : px;
        a[1] = hi ? (px * px + py * py + pz * pz) : py;

        // ---- Matrix op: D = ||p_m - c_n||^2 for 16 points x 16 spheres -----
        v8f d0 = __builtin_amdgcn_wmma_f32_16x16x4_f32(
            false, a, false, b0, (short)0, cacc0, false, false);
        v8f d1 = __builtin_amdgcn_wmma_f32_16x16x4_f32(
            false, a, false, b1, (short)0, cacc1, false, false);

        // ---- sqrt (raw v_sqrt_f32), subtract radius, min over 32 slots -----
        float mres[8];
#pragma unroll
        for (int v = 0; v < 8; v += 2) {
            float m0 = fminf(__builtin_amdgcn_sqrtf(max0(d0[v])) - r0,
                             __builtin_amdgcn_sqrtf(max0(d1[v])) - r1);
            float m1 = fminf(__builtin_amdgcn_sqrtf(max0(d0[v + 1])) - r0,
                             __builtin_amdgcn_sqrtf(max0(d1[v + 1])) - r1);
            min2_across16(m0, m1);
            mres[v] = m0;
            mres[v + 1] = m1;
        }

        // ---- Store: after the butterfly every lane of a half holds all 8 of
        // its rows. Lane 0 writes rows 0..7, lane 16 writes rows 8..15, each
        // as two b128 stores (base is a multiple of 16 -> 16B aligned).
        if (col == 0) {
            if (base + 16 <= n) {
                v4f* o = (v4f*)(out + base + 8 * hi);
                v4f lo4 = {mres[0], mres[1], mres[2], mres[3]};
                v4f hi4 = {mres[4], mres[5], mres[6], mres[7]};
                o[0] = lo4;
                o[1] = hi4;
            } else {
#pragma unroll
                for (int v = 0; v < 8; ++v) {
                    const int row = base + v + 8 * hi;
                    if (row < n) out[row] = mres[v];
                }
            }
        }
    }
}

extern "C" void kernel_launch(void* const* d_in, const int* in_sizes, int n_in,
                              void* d_out, int out_size, void* d_ws, size_t ws_size,
                              hipStream_t stream) {
    const float* points  = (const float*)d_in[0];
    const float* centers = (const float*)d_in[1];
    const float* radii   = (const float*)d_in[2];
    float* out = (float*)d_out;

    const int n = in_sizes[0] / 3;
    const int pointsPerBlock = (BLOCK_THREADS / 32) * 16 * TILES_PER_WAVE; // 2048
    const int blocks = (n + pointsPerBlock - 1) / pointsPerBlock;

    LargeSDFSurface_sdf_wmma_kernel<<<blocks, BLOCK_THREADS, 0, stream>>>(
        points, centers, radii, out, n);
}